// CounterPrimitive_74131135529265
// MI455X (gfx1250) — compile-verified
//
#include <hip/hip_runtime.h>
#include <math.h>

#define B_    4
#define L_    8192
#define D_    1024
#define K_    2
#define ROWS  (B_ * L_)        // 32768
#define EMB   64               // K * 2 * 16 freqs
#define LDSW  136              // padded LDS stride (disjoint banks for half-waves)

typedef __attribute__((ext_vector_type(2))) float v2f;
typedef __attribute__((ext_vector_type(8))) float v8f;
typedef unsigned int u32x4 __attribute__((ext_vector_type(4)));
typedef int          i32x4 __attribute__((ext_vector_type(4)));
typedef int          i32x8 __attribute__((ext_vector_type(8)));

// ---------------------------------------------------------------------------
// Kernel 1: gate projections. One wave (32 lanes) per row of x.
// ---------------------------------------------------------------------------
__global__ __launch_bounds__(256) void gates_kernel(
    const float* __restrict__ x,
    const float* __restrict__ inc_w,  const float* __restrict__ inc_b,
    const float* __restrict__ reset_w,const float* __restrict__ reset_b,
    float* __restrict__ inc_logits,
    float* __restrict__ keep_g, float* __restrict__ inc_g)
{
    const int wid  = threadIdx.x >> 5;
    const int lane = threadIdx.x & 31;
    const int row  = blockIdx.x * 8 + wid;
    const float* xr = x + (size_t)row * D_;

    float s0 = 0.f, s1 = 0.f, r0 = 0.f, r1 = 0.f;
    for (int i = lane; i < D_; i += 32) {
        float xv = xr[i];
        __builtin_prefetch(xr + i + 256, 0, 1);      // global_prefetch_b8
        s0 = fmaf(xv, inc_w[i * 2 + 0],   s0);
        s1 = fmaf(xv, inc_w[i * 2 + 1],   s1);
        r0 = fmaf(xv, reset_w[i * 2 + 0], r0);
        r1 = fmaf(xv, reset_w[i * 2 + 1], r1);
    }
    #pragma unroll
    for (int m = 16; m >= 1; m >>= 1) {
        s0 += __shfl_xor(s0, m, 32);
        s1 += __shfl_xor(s1, m, 32);
        r0 += __shfl_xor(r0, m, 32);
        r1 += __shfl_xor(r1, m, 32);
    }
    if (lane == 0) {
        float l0 = s0 + inc_b[0],   l1 = s1 + inc_b[1];
        float q0 = r0 + reset_b[0], q1 = r1 + reset_b[1];
        inc_logits[row * 2 + 0] = l0;
        inc_logits[row * 2 + 1] = l1;
        inc_g[row * 2 + 0]  = 1.f / (1.f + __expf(-l0));
        inc_g[row * 2 + 1]  = 1.f / (1.f + __expf(-l1));
        keep_g[row * 2 + 0] = 1.f - 1.f / (1.f + __expf(-q0));
        keep_g[row * 2 + 1] = 1.f - 1.f / (1.f + __expf(-q1));
    }
}

// ---------------------------------------------------------------------------
// Kernel 2: affine inclusive scan  c[t] = keep[t]*c[t-1] + inc[t].
// ---------------------------------------------------------------------------
__global__ __launch_bounds__(1024) void scan_kernel(
    const float* __restrict__ keep_g, const float* __restrict__ inc_g,
    float* __restrict__ counters)
{
    __shared__ float sk[1024];
    __shared__ float si[1024];
    __shared__ float s_carry;

    const int chain = blockIdx.x;          // 0..7
    const int b = chain >> 1, k = chain & 1;
    const int tid = threadIdx.x;
    if (tid == 0) s_carry = 0.f;
    __syncthreads();

    for (int tile = 0; tile < L_ / 1024; ++tile) {
        const int t = tile * 1024 + tid;
        const size_t idx = ((size_t)b * L_ + t) * K_ + k;
        float kk = keep_g[idx];
        float ii = inc_g[idx];
        sk[tid] = kk; si[tid] = ii;
        __syncthreads();
        #pragma unroll
        for (int off = 1; off < 1024; off <<= 1) {
            float kp = 0.f, ip = 0.f;
            if (tid >= off) { kp = sk[tid - off]; ip = si[tid - off]; }
            __syncthreads();
            if (tid >= off) { ii = ii + kk * ip; kk = kk * kp; }
            sk[tid] = kk; si[tid] = ii;
            __syncthreads();
        }
        const float carry = s_carry;
        const float c = ii + kk * carry;
        counters[idx] = c;
        __syncthreads();                   // everyone read s_carry
        if (tid == 1023) s_carry = c;
        __syncthreads();
    }
}

// ---------------------------------------------------------------------------
// Kernel 3: injection = emb(counters) @ read_w + read_b  via WMMA f32 16x16x4.
// read_w 64x128 tile staged into LDS by the Tensor Data Mover (one DMA per
// block, TENSORcnt-tracked), with TDM row padding reproducing the 136-float
// bank-conflict-free stride. emb is computed in registers from counters.
// ---------------------------------------------------------------------------
__global__ __launch_bounds__(256) void readout_kernel(
    const float* __restrict__ counters,   // (ROWS, K)
    const float* __restrict__ read_w,     // (EMB, D)
    const float* __restrict__ read_b,     // (D)
    float* __restrict__ out)              // (ROWS, D)
{
    __shared__ __align__(16) float lds_w[EMB * LDSW];

    const int m_base  = blockIdx.x * 16;
    const int n_block = blockIdx.y * 128;
    const int tid = threadIdx.x;

#if __has_builtin(__builtin_amdgcn_tensor_load_to_lds) && \
    __has_builtin(__builtin_amdgcn_s_wait_tensorcnt)
    // --- TDM staging: 64 rows x 128 cols of read_w, +8 DW pad per row ---
    if (tid < 32) {
        const unsigned long long gaddr =
            (unsigned long long)(const void*)(read_w + n_block);
        u32x4 g0;
        g0[0] = 1u;                                        // count=1 (valid D#)
        g0[1] = (unsigned)(unsigned long long)(uintptr_t)&lds_w[0]; // LDS byte addr
        g0[2] = (unsigned)(gaddr & 0xFFFFFFFFull);         // global_addr[31:0]
        g0[3] = (unsigned)((gaddr >> 32) & 0x1FFFFFFull)   // global_addr[56:32]
              | (2u << 30);                                // type=2 ("image")
        i32x8 g1;
        g1[0] = (2 << 16)        // data_size: 4 bytes
              | (1 << 20)        // pad_enable
              | (6 << 22)        // pad_interval: every 128 DWORDs
              | (7 << 25);       // pad_amount: 8 DWORDs  -> LDS row stride 136
        g1[1] = (int)(1024u << 16);   // tensor_dim0 = 1024 (bits 63:48)
        g1[2] = (int)(64u   << 16);   // tensor_dim0_hi=0 | tensor_dim1 = 64
        g1[3] = (int)(128u  << 16);   // tensor_dim1_hi=0 | tile_dim0  = 128
        g1[4] = 64;                   // tile_dim1 = 64 | tile_dim2 = 0
        g1[5] = 1024;                 // tensor_dim0_stride = 1024 elements
        g1[6] = 0;
        g1[7] = 0;
        i32x4 gz = {0, 0, 0, 0};      // groups 2/3 unused (2-D tile)
#if defined(__clang_major__) && (__clang_major__ >= 23)
        i32x8 gz8 = {0, 0, 0, 0, 0, 0, 0, 0};
        __builtin_amdgcn_tensor_load_to_lds(g0, g1, gz, gz, gz8, 0);
#else
        __builtin_amdgcn_tensor_load_to_lds(g0, g1, gz, gz, 0);
#endif
    }
#define TDM_STAGED 1
#else
    // Fallback: manual coalesced staging.
    for (int idx = tid; idx < EMB * 128; idx += 256) {
        const int kk = idx >> 7, n = idx & 127;
        lds_w[kk * LDSW + n] = read_w[kk * D_ + n_block + n];
    }
    __syncthreads();
#endif

    const int wid  = tid >> 5, lane = tid & 31;
    const int h    = lane >> 4;            // half-wave selector (K split)
    const int m    = lane & 15;            // A-matrix row within tile
    const int n16  = wid * 16;             // this wave's 16-column slice

    const int row = m_base + m;
    const float c0 = counters[row * 2 + 0];
    const float c1 = counters[row * 2 + 1];

    // Per-lane trig tables (overlapped with the TDM transfer): only the 8
    // frequencies this half-wave needs (f % 4 in {2h, 2h+1}).
    float sv[2][8], cv[2][8];
    const float step = -8.317766166719343f / 15.0f;   // -ln(4096)/15
    #pragma unroll
    for (int fi = 0; fi < 8; ++fi) {
        const int f = 4 * (fi >> 1) + 2 * h + (fi & 1);
        const float inv = __expf(step * (float)f);
        __sincosf(c0 * inv, &sv[0][fi], &cv[0][fi]);
        __sincosf(c1 * inv, &sv[1][fi], &cv[1][fi]);
    }

#ifdef TDM_STAGED
    if (tid < 32) __builtin_amdgcn_s_wait_tensorcnt(0);  // s_wait_tensorcnt 0
    __syncthreads();
#endif

    v8f acc = {};
    #pragma unroll
    for (int s = 0; s < 16; ++s) {         // K = 64 in steps of 4
        v2f a{}, bf{};
        #pragma unroll
        for (int j = 0; j < 2; ++j) {
            const int kdim = s * 4 + 2 * h + j;         // emb column index
            const int ch   = s >> 3;                    // counter channel (compile-time)
            const int fi   = 2 * (s & 3) + j;           // compressed freq index
            const float val = ((s & 7) < 4) ? sv[ch][fi] : cv[ch][fi];
            a[j]  = val;
            bf[j] = lds_w[kdim * LDSW + n16 + m];
        }
        acc = __builtin_amdgcn_wmma_f32_16x16x4_f32(
            /*neg_a=*/false, a, /*neg_b=*/false, bf,
            /*c_mod=*/(short)0, acc, /*reuse_a=*/false, /*reuse_b=*/false);
    }

    const float bias = read_b[n_block + n16 + m];
    #pragma unroll
    for (int j = 0; j < 8; ++j) {
        const int orow = m_base + j + 8 * h;            // C/D layout: M = vgpr + 8h
        out[(size_t)orow * D_ + n_block + n16 + m] = acc[j] + bias;
    }
}

// ---------------------------------------------------------------------------
extern "C" void kernel_launch(void* const* d_in, const int* in_sizes, int n_in,
                              void* d_out, int out_size, void* d_ws, size_t ws_size,
                              hipStream_t stream) {
    const float* x       = (const float*)d_in[0];
    const float* inc_w   = (const float*)d_in[1];
    const float* inc_b   = (const float*)d_in[2];
    const float* reset_w = (const float*)d_in[3];
    const float* reset_b = (const float*)d_in[4];
    const float* read_w  = (const float*)d_in[5];
    const float* read_b  = (const float*)d_in[6];

    float* out        = (float*)d_out;
    float* injection  = out;                                  // ROWS * D
    float* inc_logits = out + (size_t)ROWS * D_;              // ROWS * K
    float* counters   = inc_logits + (size_t)ROWS * K_;       // ROWS * K

    float* keep_g = (float*)d_ws;                             // ROWS * K
    float* inc_g  = keep_g + (size_t)ROWS * K_;               // ROWS * K

    gates_kernel<<<dim3(ROWS / 8), 256, 0, stream>>>(
        x, inc_w, inc_b, reset_w, reset_b, inc_logits, keep_g, inc_g);

    scan_kernel<<<dim3(B_ * K_), 1024, 0, stream>>>(keep_g, inc_g, counters);

    readout_kernel<<<dim3(ROWS / 16, D_ / 128), 256, 0, stream>>>(
        counters, read_w, read_b, injection);
}